// PennSkipGramModel_62526133895302
// MI455X (gfx1250) — compile-verified
//
#include <hip/hip_runtime.h>

// Skip-gram negative-sampling loss on MI455X (gfx1250, wave32).
// One wave = 16 batch elements. Dots computed with V_WMMA_F32_16X16X4_F32,
// taking the diagonal of 16x16 products. Memory-bound (~470MB gathers).

typedef __attribute__((ext_vector_type(2))) float v2f;
typedef __attribute__((ext_vector_type(8))) float v8f;

#define HALF_DIM 128
#define K_NEG 5
#define B_TOTAL 65536
#define WAVES_PER_BLOCK 8
#define TILES_PER_BLOCK WAVES_PER_BLOCK   // 16 batch elems per wave
#define NUM_BLOCKS (B_TOTAL / (TILES_PER_BLOCK * 16))   // 512

__device__ __forceinline__ v8f wmma_f32(v2f a, v2f b, v8f c) {
  return __builtin_amdgcn_wmma_f32_16x16x4_f32(
      /*neg_a=*/false, a, /*neg_b=*/false, b,
      /*c_mod=*/(short)0, c, /*reuse_a=*/false, /*reuse_b=*/false);
}

__device__ __forceinline__ float pick8(v8f v, int idx) {
  float r = v[0];
  r = (idx == 1) ? v[1] : r;
  r = (idx == 2) ? v[2] : r;
  r = (idx == 3) ? v[3] : r;
  r = (idx == 4) ? v[4] : r;
  r = (idx == 5) ? v[5] : r;
  r = (idx == 6) ? v[6] : r;
  r = (idx == 7) ? v[7] : r;
  return r;
}

__device__ __forceinline__ float clip10(float x) {
  return fminf(fmaxf(x, -10.0f), 10.0f);
}

// log(1 + exp(x)); inputs are pre-clipped to [-10,10] so no overflow handling
__device__ __forceinline__ float softplus_c(float x) {
  return log1pf(expf(x));
}

// One "side" (l or r): accumulate 6 matrix products (pos + 5 neg) over K=128,
// extract diagonals, return this lane's loss contribution.
__device__ __forceinline__ float side_loss(const float* __restrict__ pu,
                                           const float* __restrict__ pv,
                                           const float* const* __restrict__ pn,
                                           int lane) {
  v8f aP  = {};
  v8f aN0 = {}, aN1 = {}, aN2 = {}, aN3 = {}, aN4 = {};

  const float* p0 = pn[0];
  const float* p1 = pn[1];
  const float* p2 = pn[2];
  const float* p3 = pn[3];
  const float* p4 = pn[4];

#pragma unroll 4
  for (int k0 = 0; k0 < HALF_DIM; k0 += 4) {
    v2f a  = *(const v2f*)(pu + k0);
    v2f bp = *(const v2f*)(pv + k0);
    v2f b0 = *(const v2f*)(p0 + k0);
    v2f b1 = *(const v2f*)(p1 + k0);
    v2f b2 = *(const v2f*)(p2 + k0);
    v2f b3 = *(const v2f*)(p3 + k0);
    v2f b4 = *(const v2f*)(p4 + k0);
    aP  = wmma_f32(a, bp, aP);
    aN0 = wmma_f32(a, b0, aN0);
    aN1 = wmma_f32(a, b1, aN1);
    aN2 = wmma_f32(a, b2, aN2);
    aN3 = wmma_f32(a, b3, aN3);
    aN4 = wmma_f32(a, b4, aN4);
  }

  // Diagonal D[m][m]: VGPR (m&7) of lane m (m<8) / lane m+16 (m>=8).
  // => lanes 0..7 and 24..31 each own one batch element's scores.
  const int  idx    = lane & 7;
  const bool active = (lane < 8) || (lane >= 24);

  float sP = pick8(aP,  idx);
  float s0 = pick8(aN0, idx);
  float s1 = pick8(aN1, idx);
  float s2 = pick8(aN2, idx);
  float s3 = pick8(aN3, idx);
  float s4 = pick8(aN4, idx);

  float loss = 0.0f;
  if (active) {
    loss  = softplus_c(-clip10(sP));   // -log_sigmoid(pos_score)
    loss += softplus_c(clip10(s0));    // -log_sigmoid(-neg_score)
    loss += softplus_c(clip10(s1));
    loss += softplus_c(clip10(s2));
    loss += softplus_c(clip10(s3));
    loss += softplus_c(clip10(s4));
  }
  return loss;
}

__global__ void __launch_bounds__(256)
sg_loss_kernel(const float* __restrict__ u_l, const float* __restrict__ u_r,
               const float* __restrict__ v_l, const float* __restrict__ v_r,
               const int* __restrict__ pos_u, const int* __restrict__ pos_v_l,
               const int* __restrict__ pos_v_r, const int* __restrict__ neg_v_l,
               const int* __restrict__ neg_v_r, float* __restrict__ partials) {
  __shared__ float red[WAVES_PER_BLOCK];

  const int tid  = threadIdx.x;
  const int lane = tid & 31;
  const int wave = tid >> 5;
  const int tile = blockIdx.x * TILES_PER_BLOCK + wave;

  const int m    = lane & 15;        // batch element within tile
  const int half = lane >> 4;        // which K-half this lane loads
  const int bi   = tile * 16 + m;
  const int col  = 2 * half;         // per-lane column offset (ISA A/B layout)

  const int iu  = pos_u[bi];
  const int ivl = pos_v_l[bi];
  const int ivr = pos_v_r[bi];

  const float* pul = u_l + (size_t)iu  * HALF_DIM + col;
  const float* pur = u_r + (size_t)iu  * HALF_DIM + col;
  const float* pvl = v_l + (size_t)ivl * HALF_DIM + col;
  const float* pvr = v_r + (size_t)ivr * HALF_DIM + col;

  const float* pnl[K_NEG];
  const float* pnr[K_NEG];
#pragma unroll
  for (int k = 0; k < K_NEG; ++k) {
    pnl[k] = v_l + (size_t)neg_v_l[bi * K_NEG + k] * HALF_DIM + col;
    pnr[k] = v_r + (size_t)neg_v_r[bi * K_NEG + k] * HALF_DIM + col;
  }

  float lossLane = side_loss(pul, pvl, pnl, lane)
                 + side_loss(pur, pvr, pnr, lane);

  // wave32 butterfly reduction (inactive lanes contributed 0)
#pragma unroll
  for (int off = 16; off > 0; off >>= 1)
    lossLane += __shfl_xor(lossLane, off, 32);

  if (lane == 0) red[wave] = lossLane;
  __syncthreads();

  if (tid == 0) {
    float s = 0.0f;
#pragma unroll
    for (int w = 0; w < WAVES_PER_BLOCK; ++w) s += red[w];
    partials[blockIdx.x] = s;   // deterministic per-block partial
  }
}

__global__ void __launch_bounds__(512)
sg_reduce_kernel(const float* __restrict__ partials, float* __restrict__ out) {
  __shared__ float sm[NUM_BLOCKS];
  const int t = threadIdx.x;
  sm[t] = partials[t];
  __syncthreads();
#pragma unroll
  for (int s = NUM_BLOCKS / 2; s > 0; s >>= 1) {
    if (t < s) sm[t] += sm[t + s];
    __syncthreads();
  }
  if (t == 0) out[0] = sm[0] * (1.0f / (float)B_TOTAL);
}

extern "C" void kernel_launch(void* const* d_in, const int* in_sizes, int n_in,
                              void* d_out, int out_size, void* d_ws, size_t ws_size,
                              hipStream_t stream) {
  const float* u_l = (const float*)d_in[0];
  const float* u_r = (const float*)d_in[1];
  const float* v_l = (const float*)d_in[2];
  const float* v_r = (const float*)d_in[3];
  const int* pos_u   = (const int*)d_in[4];
  const int* pos_v_l = (const int*)d_in[5];
  const int* pos_v_r = (const int*)d_in[6];
  const int* neg_v_l = (const int*)d_in[7];
  const int* neg_v_r = (const int*)d_in[8];

  float* partials = (float*)d_ws;       // NUM_BLOCKS floats of scratch
  float* out      = (float*)d_out;      // single f32 (the mean loss)

  sg_loss_kernel<<<NUM_BLOCKS, 256, 0, stream>>>(
      u_l, u_r, v_l, v_r, pos_u, pos_v_l, pos_v_r, neg_v_l, neg_v_r, partials);
  sg_reduce_kernel<<<1, NUM_BLOCKS, 0, stream>>>(partials, out);
}